// SpGraphTransAttention_77008763617445
// MI455X (gfx1250) — compile-verified
//
#include <hip/hip_runtime.h>
#include <hip/hip_bf16.h>

#define N_IN   64
#define N_ATT  64
#define NHEAD  4
#define D_K    16

typedef float v2f __attribute__((ext_vector_type(2)));
typedef float v8f __attribute__((ext_vector_type(8)));

// ---------------------------------------------------------------------------
// Order-preserving float<->int key mapping for native integer atomic max.
// ---------------------------------------------------------------------------
__device__ __forceinline__ int f32_order_key(float f) {
    int i = __float_as_int(f);
    return (i >= 0) ? i : (i ^ 0x7FFFFFFF);
}
__device__ __forceinline__ float key_to_f32(int k) {
    int i = (k >= 0) ? k : (k ^ 0x7FFFFFFF);
    return __int_as_float(i);
}

// ---------------------------------------------------------------------------
// Kernel 1: QKV projection via V_WMMA_F32_16X16X4_F32.
// One wave computes a 16-row strip of all three projections.
// A-fragment layout (32-bit A 16x4): lane<16 holds K=k0,k0+1; lane>=16 K=k0+2,k0+3.
// B-fragment symmetric (row K striped across lanes; K split by lane half).
// C/D: acc[r] = row (r + 8*(lane>=16)), col = lane%16.
// ---------------------------------------------------------------------------
template <bool PERMUTE_HEADS>
__device__ __forceinline__ void gemm_strip(const v2f a[16],
                                           const float* __restrict__ W,
                                           const float* __restrict__ bias,
                                           float* __restrict__ out,
                                           int row0, int lane) {
    const int m     = lane & 15;
    const int khalf = (lane >> 4) * 2;   // 0 or 2
    const int rbase = (lane >> 4) * 8;   // 0 or 8

    for (int j = 0; j < 4; ++j) {        // 16-column blocks of the 64-wide output
        v8f acc = {};
        const int n = j * 16 + m;
#pragma unroll
        for (int s = 0; s < 16; ++s) {   // K = 64 in steps of 4
            const int kb = s * 4 + khalf;
            v2f bf;
            bf.x = W[kb * N_ATT + n];
            bf.y = W[(kb + 1) * N_ATT + n];
            acc = __builtin_amdgcn_wmma_f32_16x16x4_f32(
                /*neg_a=*/false, a[s], /*neg_b=*/false, bf,
                /*c_mod=*/(short)0, acc, /*reuse_a=*/false, /*reuse_b=*/false);
        }
        const float bb = bias[n];
#pragma unroll
        for (int r = 0; r < 8; ++r) {
            const int row = row0 + rbase + r;
            const float val = acc[r] + bb;
            if (PERMUTE_HEADS) {
                // column c = j*16 + m  ->  head h = j, d = m; out[n][d][h]
                out[(size_t)row * N_ATT + m * NHEAD + j] = val;
            } else {
                out[(size_t)row * N_ATT + n] = val;
            }
        }
    }
}

__global__ void __launch_bounds__(256)
qkv_wmma_kernel(const float* __restrict__ x,
                const float* __restrict__ Wq, const float* __restrict__ bq,
                const float* __restrict__ Wk, const float* __restrict__ bk,
                const float* __restrict__ Wv, const float* __restrict__ bv,
                float* __restrict__ Qo, float* __restrict__ Ko,
                float* __restrict__ Vo, int n_nodes) {
    const int lane = threadIdx.x & 31;
    const int wave = blockIdx.x * (blockDim.x >> 5) + (threadIdx.x >> 5);
    const int row0 = wave * 16;
    if (row0 >= n_nodes) return;   // wave-uniform: EXEC stays all-ones for WMMA

    const int m     = lane & 15;
    const int khalf = (lane >> 4) * 2;

    // Load the 16x64 x-tile as 16 A fragments (one GLOBAL_LOAD_B64 each).
    v2f a[16];
    const float* xrow = x + (size_t)(row0 + m) * N_IN;
#pragma unroll
    for (int s = 0; s < 16; ++s) {
        const int k0 = s * 4 + khalf;
        a[s].x = xrow[k0];
        a[s].y = xrow[k0 + 1];
    }

    gemm_strip<false>(a, Wq, bq, Qo, row0, lane);
    gemm_strip<false>(a, Wk, bk, Ko, row0, lane);
    gemm_strip<true >(a, Wv, bv, Vo, row0, lane);
}

// ---------------------------------------------------------------------------
// Kernel 2: init segment max keys (-inf) and denominators (0).
// ---------------------------------------------------------------------------
__global__ void seg_init_kernel(int* __restrict__ segmax, float* __restrict__ denom,
                                int n) {
    const int i = blockIdx.x * blockDim.x + threadIdx.x;
    if (i < n) {
        segmax[i] = INT_MIN;   // below key(-inf); acts as -infinity
        denom[i]  = 0.0f;
    }
}

// ---------------------------------------------------------------------------
// Kernel 3: per-edge scaled dot products + segment max (native int atomic max).
// ---------------------------------------------------------------------------
__global__ void __launch_bounds__(256)
edge_dot_kernel(const int* __restrict__ e0, const int* __restrict__ e1,
                const float* __restrict__ Q, const float* __restrict__ K,
                float* __restrict__ prods, int* __restrict__ segmax, int n_edges) {
    const int e = blockIdx.x * blockDim.x + threadIdx.x;
    if (e >= n_edges) return;
    const int s = e0[e];
    const int t = e1[e];
    const float4* q4 = (const float4*)(Q + (size_t)s * N_ATT);
    const float4* k4 = (const float4*)(K + (size_t)t * N_ATT);

    float acc[NHEAD] = {0.f, 0.f, 0.f, 0.f};
#pragma unroll
    for (int c = 0; c < 16; ++c) {       // 16 float4 chunks; head = c/4
        const float4 qa = q4[c];
        const float4 ka = k4[c];
        acc[c >> 2] += qa.x * ka.x + qa.y * ka.y + qa.z * ka.z + qa.w * ka.w;
    }
#pragma unroll
    for (int h = 0; h < NHEAD; ++h) {
        const float p = acc[h] * 0.25f;  // 1/sqrt(d_k) = 1/4
        prods[(size_t)e * NHEAD + h] = p;
        atomicMax(&segmax[s * NHEAD + h], f32_order_key(p));
    }
}

// ---------------------------------------------------------------------------
// Kernel 4: ex = exp(p - segmax), accumulate denominators (native f32 atomics).
// ---------------------------------------------------------------------------
__global__ void __launch_bounds__(256)
edge_exp_kernel(const int* __restrict__ e0, const float* __restrict__ prods,
                const int* __restrict__ segmax, float* __restrict__ ex_out,
                float* __restrict__ denom, int n_edges) {
    const int e = blockIdx.x * blockDim.x + threadIdx.x;
    if (e >= n_edges) return;
    const int s = e0[e];
    const float4 p = ((const float4*)prods)[e];
    float4 ex;
    ex.x = __expf(p.x - key_to_f32(segmax[s * NHEAD + 0]));
    ex.y = __expf(p.y - key_to_f32(segmax[s * NHEAD + 1]));
    ex.z = __expf(p.z - key_to_f32(segmax[s * NHEAD + 2]));
    ex.w = __expf(p.w - key_to_f32(segmax[s * NHEAD + 3]));
    ((float4*)ex_out)[e] = ex;
    atomicAdd(&denom[s * NHEAD + 0], ex.x);
    atomicAdd(&denom[s * NHEAD + 1], ex.y);
    atomicAdd(&denom[s * NHEAD + 2], ex.z);
    atomicAdd(&denom[s * NHEAD + 3], ex.w);
}

// ---------------------------------------------------------------------------
// Kernel 5: normalize attention in place.
// ---------------------------------------------------------------------------
__global__ void __launch_bounds__(256)
edge_norm_kernel(const int* __restrict__ e0, const float* __restrict__ denom,
                 float* __restrict__ att, int n_edges) {
    const int e = blockIdx.x * blockDim.x + threadIdx.x;
    if (e >= n_edges) return;
    const int s = e0[e];
    float4 ex = ((const float4*)att)[e];
    ex.x = ex.x / (denom[s * NHEAD + 0] + 1e-16f);
    ex.y = ex.y / (denom[s * NHEAD + 1] + 1e-16f);
    ex.z = ex.z / (denom[s * NHEAD + 2] + 1e-16f);
    ex.w = ex.w / (denom[s * NHEAD + 3] + 1e-16f);
    ((float4*)att)[e] = ex;
}

// ---------------------------------------------------------------------------
extern "C" void kernel_launch(void* const* d_in, const int* in_sizes, int n_in,
                              void* d_out, int out_size, void* d_ws, size_t ws_size,
                              hipStream_t stream) {
    const float* x  = (const float*)d_in[0];
    const int*   ed = (const int*)d_in[1];
    const float* Wq = (const float*)d_in[2];
    const float* bq = (const float*)d_in[3];
    const float* Wk = (const float*)d_in[4];
    const float* bk = (const float*)d_in[5];
    const float* Wv = (const float*)d_in[6];
    const float* bv = (const float*)d_in[7];

    const int n_nodes = in_sizes[0] / N_IN;
    const int n_edges = in_sizes[1] / 2;
    const int* e0 = ed;
    const int* e1 = ed + n_edges;

    // Workspace layout: Q | K | segmax keys | denom
    float* Q      = (float*)d_ws;
    float* Kt     = Q + (size_t)n_nodes * N_ATT;
    int*   segmax = (int*)(Kt + (size_t)n_nodes * N_ATT);
    float* denom  = (float*)(segmax + (size_t)n_nodes * NHEAD);

    // Output layout: attention [E,4] | v [N,16,4] | prods [E,4]
    float* att   = (float*)d_out;
    float* vout  = att + (size_t)n_edges * NHEAD;
    float* prods = vout + (size_t)n_nodes * N_ATT;

    // K1: QKV projection (WMMA f32), 1 wave per 16 rows, 8 waves per block
    {
        const int waves  = (n_nodes + 15) / 16;
        const int blocks = (waves + 7) / 8;
        qkv_wmma_kernel<<<blocks, 256, 0, stream>>>(x, Wq, bq, Wk, bk, Wv, bv,
                                                    Q, Kt, vout, n_nodes);
    }
    // K2: init segment state
    {
        const int n = n_nodes * NHEAD;
        seg_init_kernel<<<(n + 255) / 256, 256, 0, stream>>>(segmax, denom, n);
    }
    // K3: edge dot products + segment max
    edge_dot_kernel<<<(n_edges + 255) / 256, 256, 0, stream>>>(
        e0, e1, Q, Kt, prods, segmax, n_edges);
    // K4: exponentials + segment sum (ex staged in attention slot of d_out)
    edge_exp_kernel<<<(n_edges + 255) / 256, 256, 0, stream>>>(
        e0, prods, segmax, att, denom, n_edges);
    // K5: normalize
    edge_norm_kernel<<<(n_edges + 255) / 256, 256, 0, stream>>>(
        e0, denom, att, n_edges);
}